// gc_GNN_DropBlock_5841155523230
// MI455X (gfx1250) — compile-verified
//
#include <hip/hip_runtime.h>
#include <hip/hip_bf16.h>

typedef __attribute__((ext_vector_type(2))) float v2f;
typedef __attribute__((ext_vector_type(8))) float v8f;

#define HID 64
#define NCLS 10
#define NGRAPH 128

// ---------------------------------------------------------------------------
// Zero-fill (graph-capture safe replacement for hipMemsetAsync)
// ---------------------------------------------------------------------------
__global__ void fill_kernel(float* __restrict__ p, float v, unsigned n) {
    unsigned i = blockIdx.x * blockDim.x + threadIdx.x;
    if (i < n) p[i] = v;
}

// ---------------------------------------------------------------------------
// Edge scatter-add: agg[dst] += x[src].  16 threads per edge, float4 chunks.
// L2-resident (x + agg + edges < 50 MB << 192 MB L2).
// ---------------------------------------------------------------------------
__global__ __launch_bounds__(256) void scatter_add_kernel(
    const float* __restrict__ xin,
    const long long* __restrict__ src,
    const long long* __restrict__ dst,
    float* __restrict__ agg, int nEdges)
{
    unsigned t = blockIdx.x * blockDim.x + threadIdx.x;
    int e = (int)(t >> 4);
    int c = (int)(t & 15u) * 4;
    if (e >= nEdges) return;
    long long s = src[e];
    long long d = dst[e];
    const float4 v = *(const float4*)(xin + (size_t)s * HID + c);
    float* ap = agg + (size_t)d * HID + c;
    atomicAdd(ap + 0, v.x);
    atomicAdd(ap + 1, v.y);
    atomicAdd(ap + 2, v.z);
    atomicAdd(ap + 3, v.w);
}

// ---------------------------------------------------------------------------
// Node-wise GraphConv GEMM via fp32 WMMA:
//   out[n][o] = sum_k agg[n][k]*Wrel[o][k] + x[n][k]*Wroot[o][k] + b[o]
// One wave32 per 16-node tile; 4 column tiles of 16; K=64 as 16 chained
// V_WMMA_F32_16X16X4_F32 steps per operand (128 WMMAs / tile).
//
// fp32 A 16x4 layout: lane m=lane&15 holds rows; VGPR j: K = 4*ks + 2*hi + j
// fp32 B 4x16 layout: lane n=lane&15 holds cols; VGPR j: K = 4*ks + 2*hi + j
// fp32 C/D 16x16:     VGPR r: M = r + 8*hi, N = lane&15
// ---------------------------------------------------------------------------
__global__ __launch_bounds__(64) void gemm_node_kernel(
    const float* __restrict__ agg,
    const float* __restrict__ xin,
    const float* __restrict__ Wrel,
    const float* __restrict__ Wroot,
    const float* __restrict__ bias,
    float* __restrict__ out,
    int nNodes, int doRelu)
{
    const int wave = threadIdx.x >> 5;
    const int lane = threadIdx.x & 31;
    const int tile = blockIdx.x * 2 + wave;
    const int row0 = tile * 16;
    if (row0 >= nNodes) return;          // wave-uniform: EXEC stays all-1s

    const int m  = lane & 15;
    const int hi = lane >> 4;

    // Accumulators initialized with bias (broadcast along rows)
    v8f acc[4];
#pragma unroll
    for (int ct = 0; ct < 4; ++ct) {
        float bv = bias[ct * 16 + m];
#pragma unroll
        for (int r = 0; r < 8; ++r) acc[ct][r] = bv;
    }

    const float* aggRow = agg + (size_t)(row0 + m) * HID;
    const float* xRow   = xin + (size_t)(row0 + m) * HID;

    // Cache all A fragments for the 16 K-steps (2 contiguous f32 per lane each)
    v2f aRel[16], aRoot[16];
#pragma unroll
    for (int ks = 0; ks < 16; ++ks) {
        int k0 = ks * 4 + hi * 2;
        aRel[ks]  = *(const v2f*)(aggRow + k0);
        aRoot[ks] = *(const v2f*)(xRow + k0);
    }

#pragma unroll
    for (int ct = 0; ct < 4; ++ct) {
        const float* wrelRow  = Wrel  + (size_t)(ct * 16 + m) * HID;
        const float* wrootRow = Wroot + (size_t)(ct * 16 + m) * HID;
#pragma unroll
        for (int ks = 0; ks < 16; ++ks) {
            int k0 = ks * 4 + hi * 2;
            v2f brel  = *(const v2f*)(wrelRow + k0);
            acc[ct] = __builtin_amdgcn_wmma_f32_16x16x4_f32(
                false, aRel[ks], false, brel, (short)0, acc[ct], false, false);
            v2f broot = *(const v2f*)(wrootRow + k0);
            acc[ct] = __builtin_amdgcn_wmma_f32_16x16x4_f32(
                false, aRoot[ks], false, broot, (short)0, acc[ct], false, false);
        }
    }

    // Store (fused ReLU)
#pragma unroll
    for (int ct = 0; ct < 4; ++ct) {
#pragma unroll
        for (int r = 0; r < 8; ++r) {
            float v = acc[ct][r];
            if (doRelu) v = v > 0.0f ? v : 0.0f;
            int row = row0 + r + 8 * hi;
            out[(size_t)row * HID + ct * 16 + m] = v;
        }
    }
}

// ---------------------------------------------------------------------------
// Mean-pool accumulation: psum[g] += h[n], pcnt[g] += 1
// ---------------------------------------------------------------------------
__global__ __launch_bounds__(256) void pool_accum_kernel(
    const float* __restrict__ h,
    const long long* __restrict__ batch,
    float* __restrict__ psum, float* __restrict__ pcnt, int nNodes)
{
    unsigned t = blockIdx.x * blockDim.x + threadIdx.x;
    int n = (int)(t >> 4);
    int c = (int)(t & 15u) * 4;
    if (n >= nNodes) return;
    long long g = batch[n];
    const float4 v = *(const float4*)(h + (size_t)n * HID + c);
    float* pp = psum + (size_t)g * HID + c;
    atomicAdd(pp + 0, v.x);
    atomicAdd(pp + 1, v.y);
    atomicAdd(pp + 2, v.z);
    atomicAdd(pp + 3, v.w);
    if ((t & 15u) == 0) atomicAdd(pcnt + g, 1.0f);
}

// ---------------------------------------------------------------------------
// Finalize: pooled = psum / max(cnt,1);  logits = pooled @ Wlin^T + blin
// One block (64 threads) per graph; pooled staged through LDS for the dot.
// ---------------------------------------------------------------------------
__global__ __launch_bounds__(64) void finalize_kernel(
    const float* __restrict__ psum, const float* __restrict__ pcnt,
    const float* __restrict__ Wlin, const float* __restrict__ blin,
    float* __restrict__ pooled_out, float* __restrict__ cls_out)
{
    __shared__ float sp[HID];
    const int g = blockIdx.x;
    const int c = threadIdx.x;
    float cnt = pcnt[g];
    cnt = cnt > 1.0f ? cnt : 1.0f;
    float pv = psum[(size_t)g * HID + c] / cnt;
    pooled_out[(size_t)g * HID + c] = pv;
    sp[c] = pv;
    __syncthreads();
    if (c < NCLS) {
        float s = blin[c];
        const float* wr = Wlin + (size_t)c * HID;
#pragma unroll 8
        for (int k = 0; k < HID; ++k) s += sp[k] * wr[k];
        cls_out[(size_t)g * NCLS + c] = s;
    }
}

// ---------------------------------------------------------------------------
// Launch
// ---------------------------------------------------------------------------
extern "C" void kernel_launch(void* const* d_in, const int* in_sizes, int n_in,
                              void* d_out, int out_size, void* d_ws, size_t ws_size,
                              hipStream_t stream) {
    const float*     x     = (const float*)d_in[0];
    const long long* eidx  = (const long long*)d_in[1];
    const long long* batch = (const long long*)d_in[2];
    const float* W1_rel = (const float*)d_in[3];
    const float* b1     = (const float*)d_in[4];
    const float* W1_root= (const float*)d_in[5];
    const float* W2_rel = (const float*)d_in[6];
    const float* b2     = (const float*)d_in[7];
    const float* W2_root= (const float*)d_in[8];
    const float* W3_rel = (const float*)d_in[9];
    const float* b3     = (const float*)d_in[10];
    const float* W3_root= (const float*)d_in[11];
    const float* W_lin  = (const float*)d_in[12];
    const float* b_lin  = (const float*)d_in[13];

    const int nNodes = in_sizes[0] / HID;
    const int nEdges = in_sizes[1] / 2;
    const long long* src = eidx;
    const long long* dst = eidx + nEdges;

    // Workspace layout
    float* agg  = (float*)d_ws;
    float* hA   = agg + (size_t)nNodes * HID;
    float* hB   = hA  + (size_t)nNodes * HID;
    float* psum = hB  + (size_t)nNodes * HID;
    float* pcnt = psum + (size_t)NGRAPH * HID;

    float* pooled_out = (float*)d_out;                       // [128, 64]
    float* cls_out    = pooled_out + (size_t)NGRAPH * HID;   // [128, 10]

    const unsigned featN   = (unsigned)nNodes * HID;
    const unsigned fillB   = (featN + 255u) / 256u;
    const unsigned scatT   = (unsigned)nEdges * 16u;
    const unsigned scatB   = (scatT + 255u) / 256u;
    const int tiles        = (nNodes + 15) / 16;
    const int gemmB        = (tiles + 1) / 2;
    const unsigned poolT   = (unsigned)nNodes * 16u;
    const unsigned poolB   = (poolT + 255u) / 256u;

    // ---- Layer 1: agg = scatter(x); hA = relu(agg@W1rel^T + b1 + x@W1root^T)
    fill_kernel<<<fillB, 256, 0, stream>>>(agg, 0.0f, featN);
    scatter_add_kernel<<<scatB, 256, 0, stream>>>(x, src, dst, agg, nEdges);
    gemm_node_kernel<<<gemmB, 64, 0, stream>>>(agg, x, W1_rel, W1_root, b1, hA, nNodes, 1);

    // ---- Layer 2
    fill_kernel<<<fillB, 256, 0, stream>>>(agg, 0.0f, featN);
    scatter_add_kernel<<<scatB, 256, 0, stream>>>(hA, src, dst, agg, nEdges);
    gemm_node_kernel<<<gemmB, 64, 0, stream>>>(agg, hA, W2_rel, W2_root, b2, hB, nNodes, 1);

    // ---- Layer 3 (no ReLU)
    fill_kernel<<<fillB, 256, 0, stream>>>(agg, 0.0f, featN);
    scatter_add_kernel<<<scatB, 256, 0, stream>>>(hB, src, dst, agg, nEdges);
    gemm_node_kernel<<<gemmB, 64, 0, stream>>>(agg, hB, W3_rel, W3_root, b3, hA, nNodes, 0);

    // ---- Mean pool + classifier
    fill_kernel<<<(NGRAPH * HID + NGRAPH + 255) / 256, 256, 0, stream>>>(
        psum, 0.0f, (unsigned)(NGRAPH * HID + NGRAPH));
    pool_accum_kernel<<<poolB, 256, 0, stream>>>(hA, batch, psum, pcnt, nNodes);
    finalize_kernel<<<NGRAPH, 64, 0, stream>>>(psum, pcnt, W_lin, b_lin, pooled_out, cls_out);
}